// Struct_Info_70205535420505
// MI455X (gfx1250) — compile-verified
//
#include <hip/hip_runtime.h>
#include <hip/hip_bf16.h>
#include <math.h>

typedef __attribute__((ext_vector_type(2))) float v2f;
typedef __attribute__((ext_vector_type(8))) float v8f;

#define PLANE 76800   // 240*320
#define NPIX  300     // 15*20 output pixels per batch
#define CCH   16
#define KDIM  16384   // 64*16*16

// -------------------------------------------------------------------------
// Stage 1: patch conv as WMMA GEMM  [4800 x 16384] x [16384 x 16] -> feat
// grid = 300 tiles (16 GEMM rows each), block = 256 threads = 8 waves.
// Wave wv covers ci in [wv*8, wv*8+8) (K = ci*256 + kh*16 + kw). Inner
// kh/kw loops fully unrolled so all load offsets are instruction immediates
// off one per-ci base pointer -> load/WMMA-dense instruction stream.
// Partial 16x16 tiles reduced via LDS; bias + BN1 + ReLU fused in epilogue.
// feat layout: [B][N][C].
// -------------------------------------------------------------------------
__global__ __launch_bounds__(256) void conv_wmma_k(
    const float* __restrict__ x, const float* __restrict__ wgt,
    const float* __restrict__ cb,
    const float* __restrict__ g1, const float* __restrict__ b1,
    const float* __restrict__ m1, const float* __restrict__ v1,
    float* __restrict__ feat)
{
    __shared__ float red[8 * 256];
    const int tile = blockIdx.x;
    const int tid  = threadIdx.x;
    const int wv   = tid >> 5;
    const int lane = tid & 31;
    const int hi   = lane >> 4;       // K-half selector (WMMA f32 A/B layout)
    const int mn   = lane & 15;       // row m (for A) / col n (for B)

    // Per-lane A row base: output pixel gm = tile*16 + m
    const int gm = tile * 16 + mn;
    const int b  = gm / NPIX;
    const int p  = gm - b * NPIX;
    const int ho = p / 20;
    const int wo = p - ho * 20;
    // lane-constant bases; 2*hi picks the lane's K pair within each k-step
    const float* pa0 = x + (size_t)b * 64 * PLANE + ho * 16 * 320 + wo * 16 + 2 * hi;
    const float* pb0 = wgt + mn * KDIM + 2 * hi;

    v8f acc = {0.f, 0.f, 0.f, 0.f, 0.f, 0.f, 0.f, 0.f};

    for (int cc = 0; cc < 8; ++cc) {
        const int ci = wv * 8 + cc;
        const float* pa = pa0 + ci * PLANE;   // this input channel's patch rows
        const float* pb = pb0 + ci * 256;     // matching 256 weights
        if (ci < 63) __builtin_prefetch(pa + PLANE, 0, 3);
#pragma unroll
        for (int kh = 0; kh < 16; ++kh) {
#pragma unroll
            for (int s = 0; s < 4; ++s) {
                // k0 = ci*256 + kh*16 + s*4 + 2*hi ; offsets fold to immediates
                v2f av = *(const v2f*)(pa + kh * 320 + s * 4);
                v2f bv = *(const v2f*)(pb + kh * 16 + s * 4);
                acc = __builtin_amdgcn_wmma_f32_16x16x4_f32(
                    false, av, false, bv, (short)0, acc, false, false);
            }
        }
    }

    // Spill this wave's 16x16 partial C tile to LDS in (m,n) order.
    // C/D layout: vgpr r, lane l -> m = r + 8*(l>>4), n = l&15.
#pragma unroll
    for (int r = 0; r < 8; ++r) {
        const int m = r + 8 * hi;
        red[wv * 256 + m * 16 + mn] = acc[r];
    }
    __syncthreads();

    // Reduce 8 partial tiles + fused epilogue (bias, BN1, ReLU).
    float sum = 0.f;
#pragma unroll
    for (int w8 = 0; w8 < 8; ++w8) sum += red[w8 * 256 + tid];
    const int m  = tid >> 4;
    const int n  = tid & 15;
    const int go = tile * 16 + m;                 // = b*300 + p
    float val = sum + cb[n];
    val = g1[n] * (val - m1[n]) * rsqrtf(v1[n] + 1e-5f) + b1[n];
    val = fmaxf(val, 0.f);
    feat[go * CCH + n] = val;
}

// -------------------------------------------------------------------------
// Stage 2: atten = sigmoid(BN2(mean(feat) @ atten_w.T))  -> s[B][C]
// -------------------------------------------------------------------------
__global__ __launch_bounds__(256) void atten_k(
    const float* __restrict__ feat, const float* __restrict__ aw,
    const float* __restrict__ g2, const float* __restrict__ b2,
    const float* __restrict__ m2, const float* __restrict__ v2,
    float* __restrict__ s)
{
    __shared__ float mn[256];
    const int t = threadIdx.x;
    const int b = t >> 4, c = t & 15;
    float acc = 0.f;
    for (int p = 0; p < NPIX; ++p) acc += feat[(b * NPIX + p) * CCH + c];
    mn[t] = acc * (1.0f / 300.0f);
    __syncthreads();
    float a2 = 0.f;
#pragma unroll
    for (int k = 0; k < 16; ++k) a2 += mn[b * 16 + k] * aw[c * 16 + k];
    a2 = g2[c] * (a2 - m2[c]) * rsqrtf(v2[c] + 1e-5f) + b2[c];
    s[t] = 1.0f / (1.0f + expf(-a2));
}

// Stage 3: embed = feat * s  (broadcast over pixels)
__global__ void scale_k(const float* __restrict__ feat,
                        const float* __restrict__ s,
                        float* __restrict__ embed)
{
    const int i = blockIdx.x * 256 + threadIdx.x;   // 76800
    const int c = i & 15;
    const int b = i / (NPIX * CCH);
    embed[i] = feat[i] * s[b * 16 + c];
}

// -------------------------------------------------------------------------
// Stage 4: pairwise dists, stable-argsort rank selection, gather, edge linear.
// One block per (b, i): 4800 blocks x 320 threads.
// rank(j) = #{k: d[k]>d[j]} + #{k<j: d[k]==d[j]}  (matches stable argsort of -d)
// Linear layer commutes with the mean over the 16 sampled neighbors.
// kedge layout: [B][C][15][20].
// -------------------------------------------------------------------------
__global__ __launch_bounds__(320) void knn_k(
    const float* __restrict__ embed, const float* __restrict__ lw,
    const float* __restrict__ lb, float* __restrict__ kedge)
{
    __shared__ float ei[16];
    __shared__ float dist[NPIX];
    __shared__ int   knn[16];
    __shared__ float g[16];

    const int blk = blockIdx.x;
    const int b = blk / NPIX;
    const int i = blk - b * NPIX;
    const int t = threadIdx.x;
    const float* eb = embed + b * NPIX * CCH;

    if (t < 16) ei[t] = eb[i * CCH + t];
    __syncthreads();

    if (t < NPIX) {
        const float* ej = eb + t * CCH;
        float d = 0.f;
#pragma unroll
        for (int c = 0; c < 16; ++c) { float df = ei[c] - ej[c]; d += df * df; }
        dist[t] = sqrtf(d);
    }
    __syncthreads();

    if (t < NPIX) {
        const float dj = dist[t];
        int rank = 0;
        for (int k = 0; k < NPIX; ++k) {
            const float dk = dist[k];
            rank += (dk > dj) || (dk == dj && k < t);
        }
#pragma unroll
        for (int sl = 0; sl < 16; ++sl) {
            const int rt = (int)(9.375f + 18.75f * (float)sl); // exact binary
            if (rank == rt) knn[sl] = t;
        }
    }
    __syncthreads();

    if (t < 16) {                       // t = channel k
        float sum = 0.f;
#pragma unroll
        for (int sl = 0; sl < 16; ++sl) sum += eb[knn[sl] * CCH + t];
        g[t] = sum * (1.0f / 16.0f) - ei[t];
    }
    __syncthreads();

    if (t < 16) {                       // t = output channel c
        float o = lb[t];
#pragma unroll
        for (int k = 0; k < 16; ++k) o += g[k] * lw[t * 16 + k];
        const int h = i / 20, w_ = i - h * 20;
        kedge[((b * 16 + t) * 15 + h) * 20 + w_] = o;
    }
}

// -------------------------------------------------------------------------
// Bilinear 2x upsample (half-pixel centers, clipped), matching _up2x.
// -------------------------------------------------------------------------
__device__ __forceinline__ void bcoords(int i, int n, int& i0, int& i1, float& f)
{
    float src = (i + 0.5f) * 0.5f - 0.5f;
    src = fminf(fmaxf(src, 0.0f), (float)(n - 1));
    float fl = floorf(src);
    i0 = (int)fl;
    i1 = min(i0 + 1, n - 1);
    f = src - fl;
}

__global__ void up2x_k(const float* __restrict__ in, float* __restrict__ out,
                       int H, int W)
{
    const int idx = blockIdx.x * 256 + threadIdx.x;
    const int W2 = 2 * W, H2 = 2 * H;
    const int total = 256 * H2 * W2;           // B*C = 256 planes
    if (idx >= total) return;
    const int w  = idx % W2;
    const int h  = (idx / W2) % H2;
    const int bc = idx / (W2 * H2);
    const float* p = in + bc * H * W;
    int y0, y1, x0, x1; float fy, fx;
    bcoords(h, H, y0, y1, fy);
    bcoords(w, W, x0, x1, fx);
    const float l = p[y0 * W + x0] * (1.f - fy) + p[y1 * W + x0] * fy;
    const float r = p[y0 * W + x1] * (1.f - fy) + p[y1 * W + x1] * fy;
    out[idx] = l * (1.f - fx) + r * fx;
}

// Stage 6: final 2x upsample + positional encoding; out = 2*up + pe.
__global__ void final_k(const float* __restrict__ in, float* __restrict__ out)
{
    const int idx = blockIdx.x * 256 + threadIdx.x;  // 16*16*60*80
    const int H = 30, W = 40, W2 = 80, H2 = 60;
    if (idx >= 256 * H2 * W2) return;
    const int w  = idx % W2;
    const int h  = (idx / W2) % H2;
    const int bc = idx / (W2 * H2);
    const int c  = bc & 15;
    const float* p = in + bc * H * W;
    int y0, y1, x0, x1; float fy, fx;
    bcoords(h, H, y0, y1, fy);
    bcoords(w, W, x0, x1, fx);
    const float l = p[y0 * W + x0] * (1.f - fy) + p[y1 * W + x0] * fy;
    const float r = p[y0 * W + x1] * (1.f - fy) + p[y1 * W + x1] * fy;
    const float v = l * (1.f - fx) + r * fx;

    const float divs[4] = {1.0f, 0.1f, 0.01f, 0.001f};
    const float dv  = divs[c >> 2];
    const int   sel = c & 3;
    const float arg = ((sel < 2) ? (float)(w + 1) : (float)(h + 1)) * dv;
    const float pe  = (sel & 1) ? cosf(arg) : sinf(arg);
    out[idx] = 2.0f * v + pe;
}

// -------------------------------------------------------------------------
extern "C" void kernel_launch(void* const* d_in, const int* in_sizes, int n_in,
                              void* d_out, int out_size, void* d_ws, size_t ws_size,
                              hipStream_t stream)
{
    const float* x     = (const float*)d_in[0];
    const float* cw    = (const float*)d_in[1];
    const float* cb    = (const float*)d_in[2];
    const float* g1    = (const float*)d_in[3];
    const float* b1    = (const float*)d_in[4];
    const float* m1    = (const float*)d_in[5];
    const float* v1    = (const float*)d_in[6];
    const float* aw    = (const float*)d_in[7];
    const float* g2    = (const float*)d_in[8];
    const float* b2    = (const float*)d_in[9];
    const float* m2    = (const float*)d_in[10];
    const float* v2    = (const float*)d_in[11];
    const float* lw    = (const float*)d_in[12];
    const float* lb    = (const float*)d_in[13];
    float* out = (float*)d_out;

    float* ws    = (float*)d_ws;
    float* feat  = ws;                 // 16*300*16 = 76800
    float* s     = ws + 76800;         // 256
    float* embed = ws + 77056;         // 76800
    float* kedge = ws + 153856;        // 16*16*15*20 = 76800
    float* up1   = ws + 230656;        // 16*16*30*40 = 307200

    conv_wmma_k<<<300, 256, 0, stream>>>(x, cw, cb, g1, b1, m1, v1, feat);
    atten_k    <<<1,   256, 0, stream>>>(feat, aw, g2, b2, m2, v2, s);
    scale_k    <<<300, 256, 0, stream>>>(feat, s, embed);
    knn_k      <<<4800, 320, 0, stream>>>(embed, lw, lb, kedge);
    up2x_k     <<<1200, 256, 0, stream>>>(kedge, up1, 15, 20);
    final_k    <<<4800, 256, 0, stream>>>(up1, out);
}